// Discriminator_36378372997647
// MI455X (gfx1250) — compile-verified
//
#include <hip/hip_runtime.h>

// ---------------------------------------------------------------------------
// GCN forward for MI455X (gfx1250, wave32).
// GEMMs use V_WMMA_F32_16X16X4_F32 (full f32 precision, CDNA5 matrix pipe).
// Edge aggregation uses hardware global_atomic_add_f32 (non-returning).
// ---------------------------------------------------------------------------

typedef __attribute__((ext_vector_type(2))) float v2f;
typedef __attribute__((ext_vector_type(8))) float v8f;

#define FEAT 64

// ---------------------------------------------------------------------------
// Y[N,64] = X[N,64] @ W[64,64]  via f32 WMMA 16x16x4.
// One wave -> one 16-row tile, 4 accumulators cover all 64 output columns.
// A frag (16x4):  lane l holds X[row0 + l%16][kb + 2*(l/16) + v], v=0,1
// B frag (4x16):  lane l holds W[kb + 2*(l/16) + v][16*j + l%16]
// C/D (16x16):    element (m = v + 8*(l/16), n = l%16) in acc[v]
// ---------------------------------------------------------------------------
__global__ __launch_bounds__(128) void gemm64_wmma_f32(
    const float* __restrict__ X, const float* __restrict__ W,
    float* __restrict__ Y, int nRows) {
  const int wave  = blockIdx.x * (blockDim.x >> 5) + (threadIdx.x >> 5);
  const int lane  = threadIdx.x & 31;
  const int nTile = nRows >> 4;
  if (wave >= nTile) return;

  const int row0 = wave << 4;
  const int lo = lane & 15;   // M (for A) / N (for B,C)
  const int hi = lane >> 4;   // K-half selector

  v8f acc0 = {}, acc1 = {}, acc2 = {}, acc3 = {};
  const float* __restrict__ xrow = X + (size_t)(row0 + lo) * FEAT;

#pragma unroll 4
  for (int kb = 0; kb < FEAT; kb += 4) {
    const int k0 = kb + 2 * hi;
    v2f a;
    a.x = xrow[k0];
    a.y = xrow[k0 + 1];

    const float* __restrict__ w0 = W + (size_t)k0 * FEAT;      // row k0
    const float* __restrict__ w1 = w0 + FEAT;                  // row k0+1
    v2f b0, b1, b2, b3;
    b0.x = w0[lo];      b0.y = w1[lo];
    b1.x = w0[16 + lo]; b1.y = w1[16 + lo];
    b2.x = w0[32 + lo]; b2.y = w1[32 + lo];
    b3.x = w0[48 + lo]; b3.y = w1[48 + lo];

    acc0 = __builtin_amdgcn_wmma_f32_16x16x4_f32(false, a, false, b0, (short)0, acc0, false, false);
    acc1 = __builtin_amdgcn_wmma_f32_16x16x4_f32(false, a, false, b1, (short)0, acc1, false, false);
    acc2 = __builtin_amdgcn_wmma_f32_16x16x4_f32(false, a, false, b2, (short)0, acc2, false, false);
    acc3 = __builtin_amdgcn_wmma_f32_16x16x4_f32(false, a, false, b3, (short)0, acc3, false, false);
  }

  float* __restrict__ yrow = Y + (size_t)(row0 + 8 * hi) * FEAT + lo;
#pragma unroll
  for (int v = 0; v < 8; ++v) {
    float* yr = yrow + (size_t)v * FEAT;
    yr[0]  = acc0[v];
    yr[16] = acc1[v];
    yr[32] = acc2[v];
    yr[48] = acc3[v];
  }
}

// deg[i] = 1.0f  (self-loop contribution)
__global__ void init_deg(float* __restrict__ deg, int N) {
  int i = blockIdx.x * blockDim.x + threadIdx.x;
  if (i < N) deg[i] = 1.0f;
}

// deg[col[e]] += 1 over real edges
__global__ void accum_deg(const int* __restrict__ col, float* __restrict__ deg, int E) {
  int e = blockIdx.x * blockDim.x + threadIdx.x;
  if (e < E) unsafeAtomicAdd(&deg[col[e]], 1.0f);
}

// dinv[i] = rsqrt(deg[i])  (deg >= 1 guaranteed)
__global__ void finish_dinv(float* __restrict__ d, int N) {
  int i = blockIdx.x * blockDim.x + threadIdx.x;
  if (i < N) d[i] = rsqrtf(d[i]);
}

// OUT[col[e], :] += H[row[e], :] * dinv[row]*dinv[col]
// one wave per edge, float2 per lane (64 feats, 256B coalesced gather)
__global__ void gcn_aggregate(const float* __restrict__ H,
                              const int* __restrict__ rowI,
                              const int* __restrict__ colI,
                              const float* __restrict__ dinv,
                              float* __restrict__ OUT, int E) {
  int idx = blockIdx.x * blockDim.x + threadIdx.x;
  int e = idx >> 5;
  if (e >= E) return;
  int lane = idx & 31;
  int r = rowI[e];
  int c = colI[e];
  float norm = dinv[r] * dinv[c];
  const float2* __restrict__ src = (const float2*)(H + (size_t)r * FEAT);
  float2 v = src[lane];
  float* dst = OUT + (size_t)c * FEAT + lane * 2;
  unsafeAtomicAdd(dst,     v.x * norm);
  unsafeAtomicAdd(dst + 1, v.y * norm);
}

// A = leaky_relu(A + T*dinv^2 + bias)   (adds self-loop message, bias, act)
__global__ void gcn_finalize(const float* __restrict__ T, float* __restrict__ A,
                             const float* __restrict__ dinv,
                             const float* __restrict__ bias, int N) {
  int idx = blockIdx.x * blockDim.x + threadIdx.x;
  if (idx >= N * FEAT) return;
  int node = idx >> 6;
  int f = idx & (FEAT - 1);
  float d = dinv[node];
  float v = A[idx] + T[idx] * d * d + bias[f];
  A[idx] = v > 0.0f ? v : 0.01f * v;
}

// pooled[batch[n], f] += H[n, f];  cnt[batch[n]] += 1 (once per node)
__global__ void pool_accum(const float* __restrict__ H, const int* __restrict__ batch,
                           float* __restrict__ pooled, float* __restrict__ cnt, int N) {
  int idx = blockIdx.x * blockDim.x + threadIdx.x;
  if (idx >= N * FEAT) return;
  int node = idx >> 6;
  int f = idx & (FEAT - 1);
  int b = batch[node];
  unsafeAtomicAdd(&pooled[(size_t)b * FEAT + f], H[idx]);
  if (f == 0) unsafeAtomicAdd(&cnt[b], 1.0f);
}

// out[g] = dot(pooled[g,:], Wfc) / max(cnt[g],1) + bfc
__global__ void pool_final(const float* __restrict__ pooled, const float* __restrict__ cnt,
                           const float* __restrict__ Wfc, const float* __restrict__ bfc,
                           float* __restrict__ out, int G) {
  int g = blockIdx.x * blockDim.x + threadIdx.x;
  if (g >= G) return;
  const float* p = pooled + (size_t)g * FEAT;
  float s = 0.0f;
#pragma unroll
  for (int f = 0; f < FEAT; ++f) s += p[f] * Wfc[f];
  float c = cnt[g];
  c = c < 1.0f ? 1.0f : c;
  out[g] = s / c + bfc[0];
}

extern "C" void kernel_launch(void* const* d_in, const int* in_sizes, int n_in,
                              void* d_out, int out_size, void* d_ws, size_t ws_size,
                              hipStream_t stream) {
  const float* x    = (const float*)d_in[0];  // [N,64]
  const int*   eidx = (const int*)d_in[1];    // [2,E]
  const int*   bat  = (const int*)d_in[2];    // [N]
  const float* W1   = (const float*)d_in[4];
  const float* b1   = (const float*)d_in[5];
  const float* W2   = (const float*)d_in[6];
  const float* b2   = (const float*)d_in[7];
  const float* Wfc  = (const float*)d_in[8];
  const float* bfc  = (const float*)d_in[9];
  float* out = (float*)d_out;

  const int N = in_sizes[0] / FEAT;   // 100000
  const int E = in_sizes[1] / 2;      // 1000000
  const int G = out_size;             // 512

  const int* rowI = eidx;             // edge_index[0] : sources
  const int* colI = eidx + E;         // edge_index[1] : targets

  // workspace layout (floats)
  float* buf0   = (float*)d_ws;                 // N*64  (GEMM output T)
  float* buf1   = buf0 + (size_t)N * FEAT;      // N*64  (aggregate / activated A)
  float* dinv   = buf1 + (size_t)N * FEAT;      // N     (deg -> dinv in place)
  float* pooled = dinv + N;                     // G*64
  float* cnt    = pooled + (size_t)G * FEAT;    // G
  (void)ws_size; (void)n_in;

  const int TPB = 256;
  const int nNodeBlk = (N + TPB - 1) / TPB;
  const int nEdgeBlk = (E + TPB - 1) / TPB;
  const int nNF      = N * FEAT;
  const int nNFBlk   = (nNF + TPB - 1) / TPB;
  const int nAggBlk  = ((E * 32) + TPB - 1) / TPB;   // wave-per-edge
  const int nTileBlk = ((N >> 4) + 3) / 4;           // 4 waves/block, 16 rows/wave

  // --- degree / normalization -------------------------------------------
  init_deg<<<nNodeBlk, TPB, 0, stream>>>(dinv, N);
  accum_deg<<<nEdgeBlk, TPB, 0, stream>>>(colI, dinv, E);
  finish_dinv<<<nNodeBlk, TPB, 0, stream>>>(dinv, N);

  // --- layer 1 -----------------------------------------------------------
  gemm64_wmma_f32<<<nTileBlk, 128, 0, stream>>>(x, W1, buf0, N);
  hipMemsetAsync(buf1, 0, (size_t)nNF * sizeof(float), stream);
  gcn_aggregate<<<nAggBlk, TPB, 0, stream>>>(buf0, rowI, colI, dinv, buf1, E);
  gcn_finalize<<<nNFBlk, TPB, 0, stream>>>(buf0, buf1, dinv, b1, N);

  // --- layer 2 -----------------------------------------------------------
  gemm64_wmma_f32<<<nTileBlk, 128, 0, stream>>>(buf1, W2, buf0, N);
  hipMemsetAsync(buf1, 0, (size_t)nNF * sizeof(float), stream);
  gcn_aggregate<<<nAggBlk, TPB, 0, stream>>>(buf0, rowI, colI, dinv, buf1, E);
  gcn_finalize<<<nNFBlk, TPB, 0, stream>>>(buf0, buf1, dinv, b2, N);

  // --- global mean pool + fc --------------------------------------------
  hipMemsetAsync(pooled, 0, (size_t)(G * FEAT + G) * sizeof(float), stream);
  pool_accum<<<nNFBlk, TPB, 0, stream>>>(buf1, bat, pooled, cnt, N);
  pool_final<<<(G + TPB - 1) / TPB, TPB, 0, stream>>>(pooled, cnt, Wfc, bfc, out, G);
}